// Recommender_v2_decoder_59038620451155
// MI455X (gfx1250) — compile-verified
//
#include <hip/hip_runtime.h>

typedef float v2f __attribute__((ext_vector_type(2)));
typedef float v8f __attribute__((ext_vector_type(8)));

#define HDIM   1024
#define BDIM   64
#define VOCABN 50000
#define NSU 50
#define NSN 50
#define NSP 100
#define NSD 20
#define NBLK_N 16   // HDIM/64 column-chunks contributing score partials

// ---------------------------------------------------------------------------
// CDNA5 async global->LDS copy (bypasses VGPRs, tracked by ASYNCcnt).
// ---------------------------------------------------------------------------
__device__ __forceinline__ unsigned lds_off_u32(const void* p) {
    // generic (flat) pointer to LDS: low 32 bits are the LDS byte offset
    return (unsigned)(unsigned long long)p;
}
__device__ __forceinline__ void async_ld_b128(const void* lds_dst, const float* gsrc) {
    asm volatile("global_load_async_to_lds_b128 %0, %1, off"
                 :: "v"(lds_off_u32(lds_dst)), "v"(gsrc) : "memory");
}
__device__ __forceinline__ void wait_async0() {
    asm volatile("s_wait_asynccnt 0x0" ::: "memory");
}

// ---------------------------------------------------------------------------
// Attention score partials: for rows r=(s,b) of feats (S*B, nparts*H) and a
// 64-wide column chunk of W, compute sum_col tanh(X@W + bias)[r,col]*V[col].
// epart[blockIdx.y * (S*B) + r] <- partial (deterministic two-pass reduce).
// ---------------------------------------------------------------------------
__global__ __launch_bounds__(256)
void attn_score_kernel(const float* __restrict__ p0, const float* __restrict__ p1,
                       const float* __restrict__ p2, int bcast_mask, int nparts,
                       const float* __restrict__ W, const float* __restrict__ bias,
                       const float* __restrict__ V, float* __restrict__ epart)
{
    __shared__ float As[32][33];
    __shared__ float Bs[32][65];
    __shared__ float ered[32][4];

    const int m_base = blockIdx.x * 32;
    const int n_base = blockIdx.y * 64;
    const int tid  = threadIdx.x;
    const int lane = tid & 31;
    const int wave = tid >> 5;
    const int wm = wave >> 2;     // 0..1 : 16-row sub-tile
    const int wn = wave & 3;      // 0..3 : 16-col sub-tile
    const int hl = lane >> 4;     // lane half
    const int lx = lane & 15;

    v8f acc = {};
    const int K = nparts << 10;
    for (int k0 = 0; k0 < K; k0 += 32) {
        const int part = k0 >> 10;
        const int kin  = k0 & (HDIM - 1);
        const float* Ap = (part == 0) ? p0 : ((part == 1) ? p1 : p2);
        const bool bc = (bcast_mask >> part) & 1;
        {   // A tile: 32 rows x 32 k  (rows map to (s,b); bcast parts ignore s)
            const int i = tid >> 3, q = tid & 7;
            const int r = m_base + i;
            const int s = r >> 6;
            const int b = r & 63;
            async_ld_b128(&As[i][q*4],
                          Ap + ((size_t)(bc ? 0 : s) * BDIM + b) * HDIM + kin + q * 4);
        }
        #pragma unroll
        for (int t = 0; t < 2; ++t) {   // W tile: 32 k-rows x 64 cols (row-major)
            const int L = tid + t * 256;
            const int j = L >> 4, q = L & 15;
            async_ld_b128(&Bs[j][q*4], W + (size_t)(k0 + j) * HDIM + n_base + q * 4);
        }
        wait_async0();
        __syncthreads();
        const int m0 = wm * 16, n0 = wn * 16;
        #pragma unroll
        for (int ks = 0; ks < 32; ks += 4) {
            v2f a, b;
            a.x = As[m0 + lx][ks + 2*hl + 0];   // lanes 0-15: K=0,1 ; 16-31: K=2,3
            a.y = As[m0 + lx][ks + 2*hl + 1];
            b.x = Bs[ks + hl][n0 + lx];         // VGPR0: K rows 0/1 across halves
            b.y = Bs[ks + 2 + hl][n0 + lx];     // VGPR1: K rows 2/3
            acc = __builtin_amdgcn_wmma_f32_16x16x4_f32(false, a, false, b,
                                                        (short)0, acc, false, false);
        }
        __syncthreads();
    }
    // epilogue: tanh(acc + bias)*V, reduce over this block's 64 columns
    const int col = n_base + wn * 16 + lx;
    const float bb = bias[col];
    const float vv = V[col];
    #pragma unroll
    for (int v = 0; v < 8; ++v) {
        float t = tanhf(acc[v] + bb) * vv;
        t += __shfl_xor(t, 1);
        t += __shfl_xor(t, 2);
        t += __shfl_xor(t, 4);
        t += __shfl_xor(t, 8);                  // reduce within each 16-lane half
        if (lx == 0) ered[wm*16 + hl*8 + v][wn] = t;
    }
    __syncthreads();
    if (tid < 32) {
        const float s = ered[tid][0] + ered[tid][1] + ered[tid][2] + ered[tid][3];
        const int MS = gridDim.x * 32;
        epart[(size_t)blockIdx.y * MS + m_base + tid] = s;
    }
}

// e[i] = bv + sum over the 16 column-chunk partials (deterministic order)
__global__ void reduce_e_kernel(float* __restrict__ e, const float* __restrict__ epart,
                                int n, const float* __restrict__ bv)
{
    const int i = blockIdx.x * 256 + threadIdx.x;
    if (i >= n) return;
    float s = bv[0];
    for (int nb = 0; nb < NBLK_N; ++nb) s += epart[(size_t)nb * n + i];
    e[i] = s;
}

// softmax over time axis (S) per batch column; alpha written in place. 64 thr.
__global__ void softmax_time_kernel(float* __restrict__ e, int S)
{
    const int b = threadIdx.x;
    float m = -INFINITY;
    for (int s = 0; s < S; ++s) m = fmaxf(m, e[s * BDIM + b]);
    float sum = 0.f;
    for (int s = 0; s < S; ++s) {
        const float x = expf(e[s * BDIM + b] - m);
        e[s * BDIM + b] = x;
        sum += x;
    }
    const float inv = 1.f / sum;
    for (int s = 0; s < S; ++s) e[s * BDIM + b] *= inv;
}

// ctx[b,h] = sum_s alpha[s,b] * values[s,b,h]
__global__ void ctx_kernel(float* __restrict__ ctx, const float* __restrict__ alpha,
                           const float* __restrict__ values, int S)
{
    const int idx = blockIdx.x * 256 + threadIdx.x;   // b*H + h
    const int b = idx >> 10, h = idx & (HDIM - 1);
    float acc = 0.f;
    for (int s = 0; s < S; ++s)
        acc += alpha[s * BDIM + b] * values[((size_t)s * BDIM + b) * HDIM + h];
    ctx[idx] = acc;
}

// ---------------------------------------------------------------------------
// C(64,N) = A0@W0^T [+ A1@W1^T] + bias0 [+ bias1].  A: (64,H) row-major,
// W: (N,H) row-major.  W rows are staged into LDS UNtransposed (async b128
// needs contiguous 16B destinations); the transpose happens logically at
// fragment-read time.  BM=64 in ONE block so 200MB W_out streams from HBM
// exactly once (it does not fit the 192MB L2).
// ---------------------------------------------------------------------------
__global__ __launch_bounds__(256)
void gemm_bt_kernel(const float* __restrict__ A0, const float* __restrict__ A1,
                    const float* __restrict__ W0, const float* __restrict__ W1,
                    const float* __restrict__ bias0, const float* __restrict__ bias1,
                    float* __restrict__ C, int N, int nparts, int ldC)
{
    __shared__ float As[64][33];
    __shared__ float Bsn[64][33];   // Bsn[n][k] : W rows as-is

    const int n_base = blockIdx.x * 64;
    const int tid  = threadIdx.x;
    const int lane = tid & 31;
    const int wave = tid >> 5;
    const int wm = wave >> 2;     // rows wm*32 .. +31 (two 16-row sub-tiles)
    const int wn = wave & 3;
    const int hl = lane >> 4;
    const int lx = lane & 15;

    v8f acc0 = {}, acc1 = {};
    const int K = nparts << 10;
    for (int k0 = 0; k0 < K; k0 += 32) {
        const int part = k0 >> 10;
        const int kin  = k0 & (HDIM - 1);
        const float* Ap = part ? A1 : A0;
        const float* Wp = part ? W1 : W0;
        #pragma unroll
        for (int t = 0; t < 2; ++t) {   // A tile: 64 x 32
            const int L = tid + t * 256;
            const int i = L >> 3, q = L & 7;
            async_ld_b128(&As[i][q*4], Ap + (size_t)i * HDIM + kin + q * 4);
        }
        #pragma unroll
        for (int t = 0; t < 2; ++t) {   // W tile: 64 n-rows x 32 k, row-major per n
            const int L = tid + t * 256;
            const int nn = L >> 3, q = L & 7;
            if (n_base + nn < N) {
                async_ld_b128(&Bsn[nn][q*4], Wp + (size_t)(n_base + nn) * HDIM + kin + q * 4);
            } else {
                Bsn[nn][q*4+0] = 0.f; Bsn[nn][q*4+1] = 0.f;
                Bsn[nn][q*4+2] = 0.f; Bsn[nn][q*4+3] = 0.f;
            }
        }
        wait_async0();
        __syncthreads();
        const int m0 = wm * 32, n0 = wn * 16;
        #pragma unroll
        for (int ks = 0; ks < 32; ks += 4) {
            v2f a0, a1, b;
            b.x  = Bsn[n0 + lx][ks + hl];       // logical transpose on read
            b.y  = Bsn[n0 + lx][ks + 2 + hl];
            a0.x = As[m0 + lx][ks + 2*hl + 0];
            a0.y = As[m0 + lx][ks + 2*hl + 1];
            a1.x = As[m0 + 16 + lx][ks + 2*hl + 0];
            a1.y = As[m0 + 16 + lx][ks + 2*hl + 1];
            acc0 = __builtin_amdgcn_wmma_f32_16x16x4_f32(false, a0, false, b,
                                                         (short)0, acc0, false, false);
            acc1 = __builtin_amdgcn_wmma_f32_16x16x4_f32(false, a1, false, b,
                                                         (short)0, acc1, false, false);
        }
        __syncthreads();
    }
    const int col = n_base + wn * 16 + lx;
    if (col < N) {
        float bb = bias0[col];
        if (bias1) bb += bias1[col];
        const int r0 = wm * 32 + hl * 8;
        #pragma unroll
        for (int v = 0; v < 8; ++v) {
            C[(size_t)(r0 + v) * ldC + col]      = acc0[v] + bb;
            C[(size_t)(r0 + 16 + v) * ldC + col] = acc1[v] + bb;
        }
    }
}

// LSTM pointwise: gates (B,4H) in torch order i,f,g,o
__global__ void lstm_kernel(const float* __restrict__ gates, const float* __restrict__ c0,
                            float* __restrict__ h_out, float* __restrict__ c_out)
{
    const int idx = blockIdx.x * 256 + threadIdx.x;   // b*H + h
    const int b = idx >> 10, hh = idx & (HDIM - 1);
    const float* g = gates + (size_t)b * 4 * HDIM;
    const float gi = g[hh], gf = g[HDIM + hh], gg = g[2*HDIM + hh], go = g[3*HDIM + hh];
    const float si = 1.f / (1.f + expf(-gi));
    const float sf = 1.f / (1.f + expf(-gf));
    const float so = 1.f / (1.f + expf(-go));
    const float c = sf * c0[idx] + si * tanhf(gg);
    h_out[idx] = so * tanhf(c);
    c_out[idx] = c;
}

// in-place softmax over VOCAB per batch row; one block per row
__global__ __launch_bounds__(256)
void softmax_vocab_kernel(float* __restrict__ logits)
{
    __shared__ float red[256];
    float* row = logits + (size_t)blockIdx.x * VOCABN;
    const int tid = threadIdx.x;
    float m = -INFINITY;
    for (int v = tid; v < VOCABN; v += 256) m = fmaxf(m, row[v]);
    red[tid] = m; __syncthreads();
    for (int s = 128; s > 0; s >>= 1) { if (tid < s) red[tid] = fmaxf(red[tid], red[tid+s]); __syncthreads(); }
    m = red[0]; __syncthreads();
    float sum = 0.f;
    for (int v = tid; v < VOCABN; v += 256) { const float x = expf(row[v] - m); row[v] = x; sum += x; }
    red[tid] = sum; __syncthreads();
    for (int s = 128; s > 0; s >>= 1) { if (tid < s) red[tid] += red[tid+s]; __syncthreads(); }
    const float inv = 1.f / red[0];
    for (int v = tid; v < VOCABN; v += 256) row[v] *= inv;
}

extern "C" void kernel_launch(void* const* d_in, const int* in_sizes, int n_in,
                              void* d_out, int out_size, void* d_ws, size_t ws_size,
                              hipStream_t stream)
{
    (void)in_sizes; (void)n_in; (void)out_size; (void)ws_size;
    const float* user_lstm  = (const float*)d_in[0];
    const float* neigh_lstm = (const float*)d_in[2];
    const float* prod_lstm  = (const float*)d_in[4];
    const float* dec_h0     = (const float*)d_in[6];
    const float* dec_c0     = (const float*)d_in[7];
    const float* dec_inputs = (const float*)d_in[8];
    const float* W_user    = (const float*)d_in[9];
    const float* b_user    = (const float*)d_in[10];
    const float* V_user    = (const float*)d_in[11];
    const float* bv_user   = (const float*)d_in[12];
    const float* W_neigh   = (const float*)d_in[13];
    const float* b_neigh   = (const float*)d_in[14];
    const float* V_neigh   = (const float*)d_in[15];
    const float* bv_neigh  = (const float*)d_in[16];
    const float* W_over    = (const float*)d_in[17];
    const float* b_over    = (const float*)d_in[18];
    const float* V_over    = (const float*)d_in[19];
    const float* bv_over   = (const float*)d_in[20];
    const float* W_comb    = (const float*)d_in[21];
    const float* b_comb    = (const float*)d_in[22];
    const float* V_comb    = (const float*)d_in[23];
    const float* bv_comb   = (const float*)d_in[24];
    const float* W_ih      = (const float*)d_in[25];
    const float* W_hh      = (const float*)d_in[26];
    const float* b_ih      = (const float*)d_in[27];
    const float* b_hh      = (const float*)d_in[28];
    const float* W_out     = (const float*)d_in[29];
    const float* b_out     = (const float*)d_in[30];

    float* ws = (float*)d_ws;
    float* e_user    = ws;                         // 3200
    float* e_neigh   = ws + 3200;                  // 3200
    float* e_over    = ws + 6400;                  // 6400
    float* e_comb    = ws + 12800;                 // 1280
    float* ctx_user  = ws + 14080;                 // 65536 each
    float* ctx_neigh = ctx_user + 65536;
    float* ctx_over  = ctx_neigh + 65536;
    float* ctx_comb  = ctx_over + 65536;
    float* gates     = ctx_comb + 65536;           // 262144
    float* epart     = gates + 262144;             // up to 16*6400

    float* out_act = (float*)d_out;                     // (B, VOCAB)
    float* out_h   = out_act + (size_t)BDIM * VOCABN;   // (B, H)
    float* out_c   = out_h + (size_t)BDIM * HDIM;       // (B, H)

    const dim3 blk(256);

    // ---- user attention (feats = user_lstm) ----
    attn_score_kernel<<<dim3(NSU*BDIM/32, NBLK_N), blk, 0, stream>>>(
        user_lstm, nullptr, nullptr, 0x0, 1, W_user, b_user, V_user, epart);
    reduce_e_kernel<<<dim3((NSU*BDIM+255)/256), blk, 0, stream>>>(e_user, epart, NSU*BDIM, bv_user);
    softmax_time_kernel<<<1, 64, 0, stream>>>(e_user, NSU);
    ctx_kernel<<<dim3(BDIM*HDIM/256), blk, 0, stream>>>(ctx_user, e_user, user_lstm, NSU);

    // ---- neighbourhood attention ----
    attn_score_kernel<<<dim3(NSN*BDIM/32, NBLK_N), blk, 0, stream>>>(
        neigh_lstm, nullptr, nullptr, 0x0, 1, W_neigh, b_neigh, V_neigh, epart);
    reduce_e_kernel<<<dim3((NSN*BDIM+255)/256), blk, 0, stream>>>(e_neigh, epart, NSN*BDIM, bv_neigh);
    softmax_time_kernel<<<1, 64, 0, stream>>>(e_neigh, NSN);
    ctx_kernel<<<dim3(BDIM*HDIM/256), blk, 0, stream>>>(ctx_neigh, e_neigh, neigh_lstm, NSN);

    // ---- overall attention (feats = [product, user_ctx(bcast), neigh_ctx(bcast)]) ----
    attn_score_kernel<<<dim3(NSP*BDIM/32, NBLK_N), blk, 0, stream>>>(
        prod_lstm, ctx_user, ctx_neigh, 0x6, 3, W_over, b_over, V_over, epart);
    reduce_e_kernel<<<dim3((NSP*BDIM+255)/256), blk, 0, stream>>>(e_over, epart, NSP*BDIM, bv_over);
    softmax_time_kernel<<<1, 64, 0, stream>>>(e_over, NSP);
    ctx_kernel<<<dim3(BDIM*HDIM/256), blk, 0, stream>>>(ctx_over, e_over, prod_lstm, NSP);

    // ---- combine attention (feats = [overall_ctx(bcast), decoder_inputs]) ----
    attn_score_kernel<<<dim3(NSD*BDIM/32, NBLK_N), blk, 0, stream>>>(
        ctx_over, dec_inputs, nullptr, 0x1, 2, W_comb, b_comb, V_comb, epart);
    reduce_e_kernel<<<dim3((NSD*BDIM+255)/256), blk, 0, stream>>>(e_comb, epart, NSD*BDIM, bv_comb);
    softmax_time_kernel<<<1, 64, 0, stream>>>(e_comb, NSD);
    ctx_kernel<<<dim3(BDIM*HDIM/256), blk, 0, stream>>>(ctx_comb, e_comb, dec_inputs, NSD);

    // ---- LSTM step: gates = x@W_ih^T + h0@W_hh^T + b_ih + b_hh ----
    gemm_bt_kernel<<<dim3(4*HDIM/64), blk, 0, stream>>>(
        ctx_comb, dec_h0, W_ih, W_hh, b_ih, b_hh, gates, 4*HDIM, 2, 4*HDIM);
    lstm_kernel<<<dim3(BDIM*HDIM/256), blk, 0, stream>>>(gates, dec_c0, out_h, out_c);

    // ---- classify: logits = h@W_out^T + b_out, then softmax over vocab ----
    gemm_bt_kernel<<<dim3((VOCABN + 63)/64), blk, 0, stream>>>(
        out_h, nullptr, W_out, nullptr, b_out, nullptr, out_act, VOCABN, 1, VOCABN);
    softmax_vocab_kernel<<<dim3(BDIM), blk, 0, stream>>>(out_act);
}